// HGATLayer_11381663335002
// MI455X (gfx1250) — compile-verified
//
#include <hip/hip_runtime.h>
#include <math.h>

// Problem constants (match reference)
#define N_A   100000
#define N_B   50000
#define NEDGE 800000
#define HD    128          // H*D = 4*32
#define NHEAD 4
#define DHEAD 32
#define NEG_SLOPE 0.2f

typedef __attribute__((ext_vector_type(2))) float v2f;
typedef __attribute__((ext_vector_type(8))) float v8f;

// ---------------------------------------------------------------------------
// GEMM: C[nrows,128] = X[nrows,128] @ W[128,128] using V_WMMA_F32_16X16X4_F32.
// Block = 256 threads = 8 waves. Block computes a 16-row x 128-col tile;
// wave w computes the 16x16 tile at columns [16w, 16w+16).
// A tile staged in LDS with pitch 132 (bank stride 4 -> conflict-free).
// ---------------------------------------------------------------------------
__global__ __launch_bounds__(256) void gemm128_wmma(const float* __restrict__ X,
                                                    const float* __restrict__ W,
                                                    float* __restrict__ C,
                                                    int nrows) {
  __shared__ float As[16 * 132];
  const int wave = threadIdx.x >> 5;
  const int lane = threadIdx.x & 31;
  const int m0 = blockIdx.x * 16;
  if (m0 >= nrows) return;

  // cooperative load of 16x128 A tile
  for (int i = threadIdx.x; i < 16 * 128; i += 256) {
    int r = i >> 7, c = i & 127;
    As[r * 132 + c] = X[(size_t)(m0 + r) * HD + c];
  }
  __syncthreads();

  const int n0 = wave * 16;
  const int half = lane >> 4;   // 0: K,K+1  1: K+2,K+3
  const int l16 = lane & 15;
  v8f acc = {};

  for (int k = 0; k < 128; k += 4) {
    const int kk = k + half * 2;
    v2f a, b;
    // A 16x4 fragment: lane l16 holds row m0+l16, cols kk,kk+1
    a.x = As[l16 * 132 + kk];
    a.y = As[l16 * 132 + kk + 1];
    // B 4x16 fragment: lane l16 holds col n0+l16, rows kk,kk+1
    b.x = W[(size_t)kk * HD + n0 + l16];
    b.y = W[(size_t)(kk + 1) * HD + n0 + l16];
    acc = __builtin_amdgcn_wmma_f32_16x16x4_f32(false, a, false, b,
                                                (short)0, acc, false, false);
  }

  // C/D layout: VGPR g -> row g (lanes 0-15) / row g+8 (lanes 16-31), col l16
#pragma unroll
  for (int g = 0; g < 8; ++g) {
    int row = m0 + g + half * 8;
    C[(size_t)row * HD + n0 + l16] = acc[g];
  }
}

// ---------------------------------------------------------------------------
// Attention projection: out[n,h] = sum_d z[n,h*32+d] * avec[h*32+d]
// ---------------------------------------------------------------------------
__global__ void attn_proj(const float* __restrict__ z,
                          const float* __restrict__ avec,
                          float* __restrict__ out, int n) {
  int t = blockIdx.x * blockDim.x + threadIdx.x;
  if (t >= n * NHEAD) return;
  int node = t >> 2, h = t & 3;
  const float* zp = z + (size_t)node * HD + h * DHEAD;
  const float* ap = avec + h * DHEAD;
  float s = 0.f;
#pragma unroll
  for (int d = 0; d < DHEAD; ++d) s += zp[d] * ap[d];
  out[t] = s;
}

// ---------------------------------------------------------------------------
// Edge pass 1: ee = exp(leaky_relu(el[src]+er[dst])); denom[dst] += ee
// (segment-max skipped: softmax is shift-invariant; |e| is small here)
// ---------------------------------------------------------------------------
__global__ void edge_softmax_num(const int* __restrict__ src,
                                 const int* __restrict__ dst,
                                 const float* __restrict__ el,
                                 const float* __restrict__ er,
                                 float* __restrict__ ee,
                                 float* __restrict__ denom, int nE) {
  int i = blockIdx.x * blockDim.x + threadIdx.x;
  if (i >= nE) return;
  int s = src[i], d = dst[i];
#pragma unroll
  for (int h = 0; h < NHEAD; ++h) {
    float e = el[s * NHEAD + h] + er[d * NHEAD + h];
    e = (e > 0.f) ? e : NEG_SLOPE * e;
    float x = expf(e);
    ee[(size_t)i * NHEAD + h] = x;
    atomicAdd(&denom[d * NHEAD + h], x);
  }
}

// ---------------------------------------------------------------------------
// Edge pass 2: rst[dst,h,d] += z_src[src,h,d] * (ee/denom[dst,h])
// 32 lanes per edge; lane = d, loop over 4 heads (coalesced 128B per head).
// ---------------------------------------------------------------------------
__global__ void edge_aggregate(const int* __restrict__ src,
                               const int* __restrict__ dst,
                               const float* __restrict__ ee,
                               const float* __restrict__ denom,
                               const float* __restrict__ z_src,
                               float* __restrict__ rst, int nE) {
  int t = blockIdx.x * blockDim.x + threadIdx.x;
  int i = t >> 5;
  int lane = t & 31;
  if (i >= nE) return;
  int s = src[i], d = dst[i];
#pragma unroll
  for (int h = 0; h < NHEAD; ++h) {
    float alpha = ee[(size_t)i * NHEAD + h] /
                  fmaxf(denom[d * NHEAD + h], 1e-9f);
    int col = h * DHEAD + lane;
    atomicAdd(&rst[(size_t)d * HD + col],
              z_src[(size_t)s * HD + col] * alpha);
  }
}

// ---------------------------------------------------------------------------
// rst = ELU(rst + bias)
// ---------------------------------------------------------------------------
__global__ void bias_elu(float* __restrict__ rst, const float* __restrict__ bias,
                         int nelem) {
  int i = blockIdx.x * blockDim.x + threadIdx.x;
  if (i >= nelem) return;
  float v = rst[i] + bias[i & (HD - 1)];
  rst[i] = (v > 0.f) ? v : (expf(v) - 1.f);
}

// ---------------------------------------------------------------------------
// Semantic logits (fused): wlog[n,kidx] = tanh(out_k[n,:] @ sW1 + sb1) . sw2
// Block = 8 waves, one 16-row block; wave w owns hidden cols [16w,16w+16).
// Hidden tensor never materialized: tanh*sw2 reduced in-register via shuffles.
// ---------------------------------------------------------------------------
__global__ __launch_bounds__(256) void semantic_logits(
    const float* __restrict__ zk, const float* __restrict__ sW1,
    const float* __restrict__ sb1, const float* __restrict__ sw2,
    float* __restrict__ wlog, int kidx, int nrows) {
  __shared__ float As[16 * 132];
  __shared__ float red[8][16];
  const int wave = threadIdx.x >> 5;
  const int lane = threadIdx.x & 31;
  const int m0 = blockIdx.x * 16;
  if (m0 >= nrows) return;

  for (int i = threadIdx.x; i < 16 * 128; i += 256) {
    int r = i >> 7, c = i & 127;
    As[r * 132 + c] = zk[(size_t)(m0 + r) * HD + c];
  }
  __syncthreads();

  const int n0 = wave * 16;
  const int half = lane >> 4;
  const int l16 = lane & 15;
  v8f acc = {};

  for (int k = 0; k < 128; k += 4) {
    const int kk = k + half * 2;
    v2f a, b;
    a.x = As[l16 * 132 + kk];
    a.y = As[l16 * 132 + kk + 1];
    b.x = sW1[(size_t)kk * HD + n0 + l16];
    b.y = sW1[(size_t)(kk + 1) * HD + n0 + l16];
    acc = __builtin_amdgcn_wmma_f32_16x16x4_f32(false, a, false, b,
                                                (short)0, acc, false, false);
  }

  const float bcol = sb1[n0 + l16];
  const float wcol = sw2[n0 + l16];
#pragma unroll
  for (int g = 0; g < 8; ++g) {
    float v = tanhf(acc[g] + bcol) * wcol;
    // reduce over the 16 columns (lanes within each 16-lane half)
    v += __shfl_xor(v, 8, 32);
    v += __shfl_xor(v, 4, 32);
    v += __shfl_xor(v, 2, 32);
    v += __shfl_xor(v, 1, 32);
    if (l16 == 0) red[wave][half * 8 + g] = v;
  }
  __syncthreads();

  if (threadIdx.x < 16) {
    float s = 0.f;
#pragma unroll
    for (int w = 0; w < 8; ++w) s += red[w][threadIdx.x];
    wlog[(size_t)(m0 + threadIdx.x) * 2 + kidx] = s;
  }
}

// ---------------------------------------------------------------------------
// Final: softmax over K=2 semantic logits, weighted sum of the two relations.
// ---------------------------------------------------------------------------
__global__ void semantic_combine(const float* __restrict__ z0,
                                 const float* __restrict__ z1,
                                 const float* __restrict__ wlog,
                                 float* __restrict__ out, int nelem) {
  int t = blockIdx.x * blockDim.x + threadIdx.x;
  if (t >= nelem) return;
  int node = t >> 7;
  float w0 = wlog[node * 2 + 0], w1 = wlog[node * 2 + 1];
  float m = fmaxf(w0, w1);
  float e0 = expf(w0 - m), e1 = expf(w1 - m);
  float inv = 1.f / (e0 + e1);
  out[t] = (e0 * inv) * z0[t] + (e1 * inv) * z1[t];
}

// ---------------------------------------------------------------------------
extern "C" void kernel_launch(void* const* d_in, const int* in_sizes, int n_in,
                              void* d_out, int out_size, void* d_ws, size_t ws_size,
                              hipStream_t stream) {
  const float* feat_a = (const float*)d_in[0];
  const float* feat_b = (const float*)d_in[1];
  const int*   src_aa = (const int*)d_in[2];
  const int*   dst_aa = (const int*)d_in[3];
  const int*   src_ba = (const int*)d_in[4];
  const int*   dst_ba = (const int*)d_in[5];
  const float* W_aa   = (const float*)d_in[6];
  const float* al_aa  = (const float*)d_in[7];
  const float* ar_aa  = (const float*)d_in[8];
  const float* b_aa   = (const float*)d_in[9];
  const float* Ws_ba  = (const float*)d_in[10];
  const float* Wd_ba  = (const float*)d_in[11];
  const float* al_ba  = (const float*)d_in[12];
  const float* ar_ba  = (const float*)d_in[13];
  const float* b_ba   = (const float*)d_in[14];
  const float* sW1    = (const float*)d_in[15];
  const float* sb1    = (const float*)d_in[16];
  const float* sw2    = (const float*)d_in[17];
  float* out = (float*)d_out;

  // Workspace carve-out (256B aligned).
  char* p = (char*)d_ws;
  auto alloc = [&](size_t bytes) -> float* {
    float* r = (float*)p;
    p += (bytes + 255) & ~(size_t)255;
    return r;
  };
  float* z_aa   = alloc((size_t)N_A * HD * 4);     // feat_a @ W_aa  (src & dst)
  float* z_sba  = alloc((size_t)N_B * HD * 4);     // feat_b @ Ws_ba
  float* z_dba  = alloc((size_t)N_A * HD * 4);     // feat_a @ Wd_ba
  float* el_aa  = alloc((size_t)N_A * NHEAD * 4);
  float* er_aa  = alloc((size_t)N_A * NHEAD * 4);
  float* el_ba  = alloc((size_t)N_B * NHEAD * 4);
  float* er_ba  = alloc((size_t)N_A * NHEAD * 4);
  float* ee     = alloc((size_t)NEDGE * NHEAD * 4); // reused per relation
  float* denom  = alloc((size_t)N_A * NHEAD * 4);   // reused per relation
  float* rst_aa = alloc((size_t)N_A * HD * 4);
  float* rst_ba = alloc((size_t)N_A * HD * 4);
  float* wlog   = alloc((size_t)N_A * 2 * 4);

  const int TPB = 256;

  // --- Phase 1: feature projections (WMMA f32) ---
  gemm128_wmma<<<N_A / 16, TPB, 0, stream>>>(feat_a, W_aa,  z_aa,  N_A);
  gemm128_wmma<<<N_B / 16, TPB, 0, stream>>>(feat_b, Ws_ba, z_sba, N_B);
  gemm128_wmma<<<N_A / 16, TPB, 0, stream>>>(feat_a, Wd_ba, z_dba, N_A);

  attn_proj<<<(N_A * NHEAD + TPB - 1) / TPB, TPB, 0, stream>>>(z_aa,  al_aa, el_aa, N_A);
  attn_proj<<<(N_A * NHEAD + TPB - 1) / TPB, TPB, 0, stream>>>(z_aa,  ar_aa, er_aa, N_A);
  attn_proj<<<(N_B * NHEAD + TPB - 1) / TPB, TPB, 0, stream>>>(z_sba, al_ba, el_ba, N_B);
  attn_proj<<<(N_A * NHEAD + TPB - 1) / TPB, TPB, 0, stream>>>(z_dba, ar_ba, er_ba, N_A);

  // --- Phase 2: relation a->a edge softmax + aggregation ---
  hipMemsetAsync(denom,  0, (size_t)N_A * NHEAD * 4, stream);
  hipMemsetAsync(rst_aa, 0, (size_t)N_A * HD * 4, stream);
  edge_softmax_num<<<(NEDGE + TPB - 1) / TPB, TPB, 0, stream>>>(
      src_aa, dst_aa, el_aa, er_aa, ee, denom, NEDGE);
  edge_aggregate<<<((size_t)NEDGE * 32 + TPB - 1) / TPB, TPB, 0, stream>>>(
      src_aa, dst_aa, ee, denom, z_aa, rst_aa, NEDGE);

  // --- Phase 3: relation b->a (reuses ee/denom; stream-ordered) ---
  hipMemsetAsync(denom,  0, (size_t)N_A * NHEAD * 4, stream);
  hipMemsetAsync(rst_ba, 0, (size_t)N_A * HD * 4, stream);
  edge_softmax_num<<<(NEDGE + TPB - 1) / TPB, TPB, 0, stream>>>(
      src_ba, dst_ba, el_ba, er_ba, ee, denom, NEDGE);
  edge_aggregate<<<((size_t)NEDGE * 32 + TPB - 1) / TPB, TPB, 0, stream>>>(
      src_ba, dst_ba, ee, denom, z_sba, rst_ba, NEDGE);

  // --- Phase 4: bias + ELU ---
  bias_elu<<<(N_A * HD + TPB - 1) / TPB, TPB, 0, stream>>>(rst_aa, b_aa, N_A * HD);
  bias_elu<<<(N_A * HD + TPB - 1) / TPB, TPB, 0, stream>>>(rst_ba, b_ba, N_A * HD);

  // --- Phase 5: semantic attention (fused WMMA + tanh + dot) ---
  semantic_logits<<<N_A / 16, TPB, 0, stream>>>(rst_aa, sW1, sb1, sw2, wlog, 0, N_A);
  semantic_logits<<<N_A / 16, TPB, 0, stream>>>(rst_ba, sW1, sb1, sw2, wlog, 1, N_A);
  semantic_combine<<<(N_A * HD + TPB - 1) / TPB, TPB, 0, stream>>>(
      rst_aa, rst_ba, wlog, out, N_A * HD);
}